// FieldFormer_MHTAP_53884659695851
// MI455X (gfx1250) — compile-verified
//
#include <hip/hip_runtime.h>
#include <math.h>

// ---------------- WMMA f32 16x16x4 helpers (CDNA5 / gfx1250, wave32) --------
typedef __attribute__((ext_vector_type(2))) float v2f;
typedef __attribute__((ext_vector_type(8))) float v8f;

__device__ __forceinline__ v8f wmma4(v2f a, v2f b, v8f c) {
  // D = A(16x4) * B(4x16) + C(16x16), fp32
  return __builtin_amdgcn_wmma_f32_16x16x4_f32(
      /*neg_a=*/false, a, /*neg_b=*/false, b,
      /*c_mod=*/(short)0, c, /*reuse_a=*/false, /*reuse_b=*/false);
}

__device__ __forceinline__ v2f ld2(const float* p) {
  return *reinterpret_cast<const v2f*>(p);
}

// Unconditional C/D tile store (padded output buffers).
// VGPR v holds row (v) for lanes 0-15 and row (v+8) for lanes 16-31.
__device__ __forceinline__ void store_tile(float* C, int ldc, int rBase, int cBase,
                                           int lane, v8f acc) {
  int col = cBase + (lane & 15);
  int r0 = rBase + ((lane >> 4) << 3);
#pragma unroll
  for (int v = 0; v < 8; ++v) C[(r0 + v) * ldc + col] = acc[v];
}

#define N_PATCH 216
#define DIM 125
#define HEADS 8
#define HEAD_DIM 64
#define QK 512
#define MP 224      // padded patch rows (14 tiles)
#define KP 128      // padded projection K

// ---------- Stage 0: zero-pad operands into workspace -----------------------
// xp[224][128], Wqp[512][128], Wkp[512][128], W1p[32][72]
__global__ void pad_kernel(const float* __restrict__ x, const float* __restrict__ Wq,
                           const float* __restrict__ Wk, const float* __restrict__ W1,
                           float* __restrict__ xp, float* __restrict__ Wqp,
                           float* __restrict__ Wkp, float* __restrict__ W1p) {
  int i = blockIdx.x * blockDim.x + threadIdx.x;
  if (i < 28672) {                       // xp
    int r = i >> 7, k = i & 127;
    xp[i] = (r < N_PATCH && k < DIM) ? x[r * DIM + k] : 0.f;
  } else if (i < 94208) {                // Wqp
    int j = i - 28672;
    int r = j >> 7, k = j & 127;
    Wqp[j] = (k < DIM) ? Wq[r * DIM + k] : 0.f;
  } else if (i < 159744) {               // Wkp
    int j = i - 94208;
    int r = j >> 7, k = j & 127;
    Wkp[j] = (k < DIM) ? Wk[r * DIM + k] : 0.f;
  } else if (i < 162048) {               // W1p [32][72]
    int j = i - 159744;
    int r = j / 72;
    W1p[j] = (r < 20) ? W1[j] : 0.f;
  }
}

// ---------- Stage 1: q = x @ Wq^T, k = x @ Wk^T (padded: 224x128x512) -------
// one wave per 16x16 tile; 2 tensors * 14 * 32 = 896 waves; branch-free loop
__global__ void proj_kernel(const float* __restrict__ xp, const float* __restrict__ Wqp,
                            const float* __restrict__ Wkp, float* __restrict__ q,
                            float* __restrict__ k) {
  int wid = (blockIdx.x * blockDim.x + threadIdx.x) >> 5;
  int lane = threadIdx.x & 31;
  int tensor = wid / 448;
  int rem = wid % 448;
  int mBase = (rem / 32) * 16, nBase = (rem % 32) * 16;
  const float* W = tensor ? Wkp : Wqp;
  float* out = tensor ? k : q;
  int koff = (lane >> 4) << 1;                       // 0 or 2
  const float* aBase = xp + (mBase + (lane & 15)) * KP + koff;
  const float* bBase = W + (nBase + (lane & 15)) * KP + koff;
  v8f acc = {};
#pragma unroll 4
  for (int kb = 0; kb < KP; kb += 4) {
    v2f a = ld2(aBase + kb);
    v2f b = ld2(bBase + kb);
    acc = wmma4(a, b, acc);
  }
  store_tile(out, QK, mBase, nBase, lane, acc);
}

// ---------- Stage 2a: BN stats per (tensor, head) over 216*64 elems ---------
__global__ void bnstats_kernel(const float* __restrict__ q, const float* __restrict__ k,
                               float* __restrict__ stats) {
  __shared__ float s1[256], s2[256];
  int b = blockIdx.x, t = threadIdx.x;
  int tensor = b >> 3, h = b & 7;
  const float* src = tensor ? k : q;
  float s = 0.f, ss = 0.f;
  for (int i = t; i < N_PATCH * HEAD_DIM; i += 256) {
    int n = i >> 6, d = i & 63;
    float v = src[n * QK + h * HEAD_DIM + d];
    s += v; ss += v * v;
  }
  s1[t] = s; s2[t] = ss; __syncthreads();
  for (int o = 128; o > 0; o >>= 1) {
    if (t < o) { s1[t] += s1[t + o]; s2[t] += s2[t + o]; }
    __syncthreads();
  }
  if (t == 0) {
    float inv_n = 1.f / (float)(N_PATCH * HEAD_DIM);
    float mean = s1[0] * inv_n;
    float var = s2[0] * inv_n - mean * mean;
    stats[b * 2] = mean;
    stats[b * 2 + 1] = rsqrtf(var + 1e-5f);
  }
}

// ---------- Stage 2b: BN apply + L2-normalize each 64-row (one wave/row) ----
// covers all 224 padded rows (pad rows are finite garbage, never consumed).
// fold 1/sqrt(125) into Qn. Output layout [head][224][64].
__global__ void bnl2_kernel(const float* __restrict__ q, const float* __restrict__ k,
                            const float* __restrict__ stats, const float* __restrict__ bw,
                            const float* __restrict__ bb, float* __restrict__ Qn,
                            float* __restrict__ Kn) {
  int wid = (blockIdx.x * blockDim.x + threadIdx.x) >> 5;
  int lane = threadIdx.x & 31;
  int tensor = wid / (HEADS * MP);
  int rr = wid % (HEADS * MP);
  int h = rr / MP, n = rr % MP;
  const float* src = tensor ? k : q;
  float* dst = tensor ? Kn : Qn;
  float mean = stats[(tensor * 8 + h) * 2];
  float istd = stats[(tensor * 8 + h) * 2 + 1];
  float w = bw[h], bi = bb[h];
  float v0 = (src[n * QK + h * HEAD_DIM + lane]      - mean) * istd * w + bi;
  float v1 = (src[n * QK + h * HEAD_DIM + lane + 32] - mean) * istd * w + bi;
  float ss = v0 * v0 + v1 * v1;
#pragma unroll
  for (int m = 16; m > 0; m >>= 1) ss += __shfl_xor(ss, m, 32);
  float scale = 1.f / fmaxf(sqrtf(ss), 1e-12f);
  if (tensor == 0) scale *= 0.08944271909999159f;  // 1/sqrt(125)
  dst[h * (MP * HEAD_DIM) + n * HEAD_DIM + lane]      = v0 * scale;
  dst[h * (MP * HEAD_DIM) + n * HEAD_DIM + lane + 32] = v1 * scale;
}

// ---------- Stage 3: S[h][m][n] = Kn[h][m]·Qn[h][n] (padded 224x224, K=64) --
// 8 heads * 14 * 14 = 1568 waves; branch-free
__global__ void scores_kernel(const float* __restrict__ Qn, const float* __restrict__ Kn,
                              float* __restrict__ S) {
  int wid = (blockIdx.x * blockDim.x + threadIdx.x) >> 5;
  int lane = threadIdx.x & 31;
  int h = wid / 196;
  int rem = wid % 196;
  int mBase = (rem / 14) * 16, nBase = (rem % 14) * 16;
  const float* Kh = Kn + h * (MP * HEAD_DIM);
  const float* Qh = Qn + h * (MP * HEAD_DIM);
  int koff = (lane >> 4) << 1;
  const float* aBase = Kh + (mBase + (lane & 15)) * HEAD_DIM + koff;
  const float* bBase = Qh + (nBase + (lane & 15)) * HEAD_DIM + koff;
  v8f acc = {};
#pragma unroll 4
  for (int kb = 0; kb < HEAD_DIM; kb += 4) {
    v2f a = ld2(aBase + kb);
    v2f b = ld2(bBase + kb);
    acc = wmma4(a, b, acc);
  }
  store_tile(S + h * MP * MP, MP, mBase, nBase, lane, acc);
}

// ---------- Stage 4: sparsemax over n for each column (h,m); 1728 blocks ----
__global__ void sparsemax_kernel(const float* __restrict__ S, float* __restrict__ att) {
  __shared__ float zs[256];
  __shared__ float cs[256];
  __shared__ float tauS;
  int t = threadIdx.x;
  int col = blockIdx.x;  // h*216 + m
  int h = col / N_PATCH, m = col % N_PATCH;
  const float* colp = S + h * MP * MP + m * MP;
  float zorig = (t < N_PATCH) ? colp[t] : 0.f;
  zs[t] = (t < N_PATCH) ? zorig : -3.0e38f;
  __syncthreads();
  // bitonic sort, descending, 256 elements
  for (int k2 = 2; k2 <= 256; k2 <<= 1) {
    for (int j = k2 >> 1; j > 0; j >>= 1) {
      int ixj = t ^ j;
      if (ixj > t) {
        float a = zs[t], b = zs[ixj];
        bool blockDesc = ((t & k2) == 0);
        bool doSwap = blockDesc ? (a < b) : (a > b);
        if (doSwap) { zs[t] = b; zs[ixj] = a; }
      }
      __syncthreads();
    }
  }
  // inclusive prefix sum (Hillis-Steele)
  cs[t] = zs[t];
  __syncthreads();
  for (int off = 1; off < 256; off <<= 1) {
    float add = (t >= off) ? cs[t - off] : 0.f;
    __syncthreads();
    cs[t] += add;
    __syncthreads();
  }
  // support set size k* and threshold tau
  bool sup = (t < N_PATCH) && (1.f + (float)(t + 1) * zs[t] > cs[t]);
  int kstar = __syncthreads_count(sup);
  if (t == kstar - 1) tauS = (cs[t] - 1.f) / (float)kstar;
  __syncthreads();
  if (t < N_PATCH) {
    float a = zorig - tauS;
    att[t * (HEADS * N_PATCH) + col] = a > 0.f ? a : 0.f;
  }
}

// ---------- Stage 5: permute att -> x4[72][72][72] ---------------------------
__global__ void gather_kernel(const float* __restrict__ att, float* __restrict__ x4) {
  int o = blockIdx.x * blockDim.x + threadIdx.x;
  if (o >= 373248) return;
  int K = o % 72, J = (o / 72) % 72, I = o / 5184;
  int d0 = I / 36, d3 = (I / 6) % 6, d6 = I % 6;
  int d1 = J / 36, d4 = (J / 6) % 6, d7 = J % 6;
  int d2 = K / 36, d5 = (K / 6) % 6, d8 = K % 6;
  int f = d0 * 186624 + d1 * 93312 + d2 * 46656 + d3 * 7776 + d4 * 1296 +
          d5 * 216 + d6 * 36 + d7 * 6 + d8;
  x4[o] = att[f];
}

// ---------- Stage 6a: t1[32][5184] = W1p[32][72] @ x4[72][5184] (WMMA) ------
// 2 mTiles * 324 nTiles = 648 waves; branch-free
__global__ void tcl1_kernel(const float* __restrict__ W1p, const float* __restrict__ x4,
                            float* __restrict__ t1) {
  int wid = (blockIdx.x * blockDim.x + threadIdx.x) >> 5;
  int lane = threadIdx.x & 31;
  int mBase = (wid / 324) * 16, nBase = (wid % 324) * 16;
  int koff = (lane >> 4) << 1;
  const float* aBase = W1p + (mBase + (lane & 15)) * 72 + koff;
  const float* bBase = x4 + koff * 5184 + nBase + (lane & 15);
  v8f acc = {};
#pragma unroll 4
  for (int kb = 0; kb < 72; kb += 4) {
    v2f a = ld2(aBase + kb);
    v2f b;
    b.x = bBase[kb * 5184];
    b.y = bBase[kb * 5184 + 5184];
    acc = wmma4(a, b, acc);
  }
  store_tile(t1, 5184, mBase, nBase, lane, acc);
}

// ---------- Stage 6b: t2[a][c][K] = sum_J W2[c][J] * t1[a][J][K] ------------
__global__ void tcl2_kernel(const float* __restrict__ W2, const float* __restrict__ t1,
                            float* __restrict__ t2) {
  int i = blockIdx.x * blockDim.x + threadIdx.x;
  if (i >= 20 * 20 * 72) return;
  int K = i % 72, c = (i / 72) % 20, a = i / 1440;
  float s = 0.f;
  for (int J = 0; J < 72; ++J) s += W2[c * 72 + J] * t1[a * 5184 + J * 72 + K];
  t2[i] = s;
}

// ---------- Stage 6c: out[a][c][d] = tanh(sum_K W3[d][K] * t2[a][c][K]) -----
__global__ void tcl3_kernel(const float* __restrict__ W3, const float* __restrict__ t2,
                            float* __restrict__ out) {
  int i = blockIdx.x * blockDim.x + threadIdx.x;
  if (i >= 8000) return;
  int d = i % 20, c = (i / 20) % 20, a = i / 400;
  float s = 0.f;
  for (int K = 0; K < 72; ++K) s += W3[d * 72 + K] * t2[a * 1440 + c * 72 + K];
  out[i] = tanhf(s);
}

extern "C" void kernel_launch(void* const* d_in, const int* in_sizes, int n_in,
                              void* d_out, int out_size, void* d_ws, size_t ws_size,
                              hipStream_t stream) {
  const float* patches = (const float*)d_in[0];  // [216,125]
  const float* Wq = (const float*)d_in[1];       // [512,125]
  const float* Wk = (const float*)d_in[2];
  const float* bw = (const float*)d_in[3];       // [8]
  const float* bb = (const float*)d_in[4];
  const float* W1 = (const float*)d_in[5];       // [20,72]
  const float* W2 = (const float*)d_in[6];
  const float* W3 = (const float*)d_in[7];
  float* out = (float*)d_out;          // [8000 tanh(x5)] ++ [373248 att]
  float* att = out + 8000;

  float* ws = (float*)d_ws;
  float* xp    = ws;               //  28672  [224][128]
  float* Wqp   = ws + 28672;       //  65536  [512][128]
  float* Wkp   = ws + 94208;       //  65536
  float* W1p   = ws + 159744;      //   2304  [32][72]
  float* q     = ws + 162048;      // 114688  [224][512]
  float* k     = ws + 276736;      // 114688
  float* Qn    = ws + 391424;      // 114688  [8][224][64]
  float* Kn    = ws + 506112;      // 114688
  float* S     = ws + 620800;      // 401408  [8][224][224]
  float* x4    = ws + 1022208;     // 373248  [72][72][72]
  float* t1    = ws + 1395456;     // 165888  [32][5184]
  float* t2    = ws + 1561344;     //  28800  [20][20][72]
  float* stats = ws + 1590144;     //     32

  pad_kernel<<<633, 256, 0, stream>>>(patches, Wq, Wk, W1, xp, Wqp, Wkp, W1p);
  proj_kernel<<<112, 256, 0, stream>>>(xp, Wqp, Wkp, q, k);             // 896 waves
  bnstats_kernel<<<16, 256, 0, stream>>>(q, k, stats);
  bnl2_kernel<<<448, 256, 0, stream>>>(q, k, stats, bw, bb, Qn, Kn);    // 3584 waves
  scores_kernel<<<196, 256, 0, stream>>>(Qn, Kn, S);                    // 1568 waves
  sparsemax_kernel<<<1728, 256, 0, stream>>>(S, att);
  gather_kernel<<<1458, 256, 0, stream>>>(att, x4);
  tcl1_kernel<<<81, 256, 0, stream>>>(W1p, x4, t1);                     // 648 waves
  tcl2_kernel<<<113, 256, 0, stream>>>(W2, t1, t2);
  tcl3_kernel<<<32, 256, 0, stream>>>(W3, t2, out);
}